// MyGaussianModel_45414984188103
// MI455X (gfx1250) — compile-verified
//
#include <hip/hip_runtime.h>
#include <hip/hip_bf16.h>
#include <math.h>
#include <stdint.h>

// ---------------------------------------------------------------------------
// Gaussian-avatar binding kernel for MI455X (gfx1250).
// Phase 1: per-face frame/quat/scale table into d_ws (9,976 faces, 80B/face).
// Phase 2: 2M-gaussian streaming kernel. Memory-bound (~232MB @ 23.3TB/s).
// gfx1250-specific paths used:
//   * global_prefetch_b8 on the gathered face record (randomly indexed,
//     L2-resident ~780KB table)
//   * non-temporal (TH_NT) hints on all streamed loads
//   * per-wave output staging in LDS + GLOBAL_STORE_ASYNC_FROM_LDS_B128/B64:
//     each wave emits 4 async ops moving a contiguous 1792B tile LDS->HBM
//     (ASYNCcnt-tracked), instead of 7 strided per-lane b64 stores.
// ---------------------------------------------------------------------------

#define FACE_STRIDE 20   // floats per face record (80 bytes): R(9) C(3) fs(1) q(4) pad(3)

typedef float v2f __attribute__((ext_vector_type(2)));
typedef float v4f __attribute__((ext_vector_type(4)));

__device__ __forceinline__ float inv_sqrt_clip(float d) {
    // matches x / sqrt(clip(d, 1e-20))
    return 1.0f / sqrtf(fmaxf(d, 1e-20f));
}

// ------------------------- Phase 1: per-face table -------------------------
__global__ void face_table_kernel(const float* __restrict__ verts,
                                  const int*   __restrict__ faces,
                                  float*       __restrict__ ftab,
                                  int n_faces) {
    int f = blockIdx.x * blockDim.x + threadIdx.x;
    if (f >= n_faces) return;

    int i0 = faces[f * 3 + 0];
    int i1 = faces[f * 3 + 1];
    int i2 = faces[f * 3 + 2];

    float v0x = verts[i0 * 3 + 0], v0y = verts[i0 * 3 + 1], v0z = verts[i0 * 3 + 2];
    float v1x = verts[i1 * 3 + 0], v1y = verts[i1 * 3 + 1], v1z = verts[i1 * 3 + 2];
    float v2x = verts[i2 * 3 + 0], v2y = verts[i2 * 3 + 1], v2z = verts[i2 * 3 + 2];

    const float third = 1.0f / 3.0f;
    float cx = (v0x + v1x + v2x) * third;
    float cy = (v0y + v1y + v2y) * third;
    float cz = (v0z + v1z + v2z) * third;

    // a0 = safe_normalize(v1 - v0)
    float e1x = v1x - v0x, e1y = v1y - v0y, e1z = v1z - v0z;
    float d1  = e1x * e1x + e1y * e1y + e1z * e1z;
    float in1 = inv_sqrt_clip(d1);
    float a0x = e1x * in1, a0y = e1y * in1, a0z = e1z * in1;

    // c = cross(a0, v2 - v0)
    float e2x = v2x - v0x, e2y = v2y - v0y, e2z = v2z - v0z;
    float cxv = a0y * e2z - a0z * e2y;
    float cyv = a0z * e2x - a0x * e2z;
    float czv = a0x * e2y - a0y * e2x;

    // a1 = safe_normalize(c)
    float dcv = cxv * cxv + cyv * cyv + czv * czv;
    float inc = inv_sqrt_clip(dcv);
    float a1x = cxv * inc, a1y = cyv * inc, a1z = czv * inc;

    // a2 = -safe_normalize(cross(a1, a0))
    float tx = a1y * a0z - a1z * a0y;
    float ty = a1z * a0x - a1x * a0z;
    float tz = a1x * a0y - a1y * a0x;
    float dt = tx * tx + ty * ty + tz * tz;
    float it = inv_sqrt_clip(dt);
    float a2x = -tx * it, a2y = -ty * it, a2z = -tz * it;

    // R = stack([a0,a1,a2], axis=-1): columns are a0,a1,a2 -> row i = (a0[i],a1[i],a2[i])
    float r00 = a0x, r01 = a1x, r02 = a2x;
    float r10 = a0y, r11 = a1y, r12 = a2y;
    float r20 = a0z, r21 = a1z, r22 = a2z;

    // face scaling
    float s0 = sqrtf(d1);
    float s1 = fabsf(cxv * a1x + cyv * a1y + czv * a1z);
    float fs = 0.5f * (s0 + s1);

    // rotmat -> unit quat (xyzw), argmax with first-max tie rule
    float tr = r00 + r11 + r22;
    int   choice = 0;
    float best   = r00;
    if (r11 > best) { best = r11; choice = 1; }
    if (r22 > best) { best = r22; choice = 2; }
    if (tr  > best) { best = tr;  choice = 3; }

    float qx, qy, qz, qw;
    if (choice == 0) {
        qx = 1.0f + r00 - r11 - r22; qy = r10 + r01; qz = r20 + r02; qw = r21 - r12;
    } else if (choice == 1) {
        qx = r10 + r01; qy = 1.0f - r00 + r11 - r22; qz = r21 + r12; qw = r02 - r20;
    } else if (choice == 2) {
        qx = r20 + r02; qy = r21 + r12; qz = 1.0f - r00 - r11 + r22; qw = r10 - r01;
    } else {
        qx = r21 - r12; qy = r02 - r20; qz = r10 - r01; qw = 1.0f + tr;
    }
    float qn = 1.0f / sqrtf(qx * qx + qy * qy + qz * qz + qw * qw);
    qx *= qn; qy *= qn; qz *= qn; qw *= qn;
    // reference re-normalizes the gathered face quat with +1e-12 eps; fold it in here
    float n2  = sqrtf(qw * qw + qx * qx + qy * qy + qz * qz);
    float in2 = 1.0f / (n2 + 1e-12f);

    float* o = ftab + (size_t)f * FACE_STRIDE;
    o[0] = r00; o[1] = r01; o[2] = r02;
    o[3] = r10; o[4] = r11; o[5] = r12;
    o[6] = r20; o[7] = r21; o[8] = r22;
    o[9] = cx;  o[10] = cy; o[11] = cz;
    o[12] = fs;
    o[13] = qw * in2; o[14] = qx * in2; o[15] = qy * in2; o[16] = qz * in2;
    o[17] = 0.0f; o[18] = 0.0f; o[19] = 0.0f;
}

// ------------------------- Phase 2: per-gaussian ---------------------------
__global__ void __launch_bounds__(256)
gaussian_kernel(const float* __restrict__ xyz,
                const float* __restrict__ opac,
                const float* __restrict__ scal,
                const float* __restrict__ rotr,
                const float* __restrict__ fdc,
                const int*   __restrict__ binding,
                const float* __restrict__ ftab,
                float*       __restrict__ out,
                int n_gauss) {
    // per-block output staging: 256 rows x 14 floats = 14336 B
    __shared__ __align__(16) float smem[256 * 14];

    const unsigned tid       = threadIdx.x;
    const unsigned blockBase = blockIdx.x * 256u;
    const unsigned g         = blockBase + tid;
    const bool     fullBlock = (blockBase + 256u) <= (unsigned)n_gauss;

    if (!fullBlock && g >= (unsigned)n_gauss) return;

    // Gather index first, then prefetch the face record (2 cachelines) so the
    // L2-resident gather overlaps the streaming loads below.
    int b = __builtin_nontemporal_load(binding + g);
    const float* fr = ftab + (size_t)b * FACE_STRIDE;
    __builtin_prefetch(fr, 0, 0);
    __builtin_prefetch(fr + 16, 0, 0);

    // Streaming (non-temporal) per-gaussian inputs.
    size_t g3 = (size_t)g * 3;
    float px = __builtin_nontemporal_load(xyz + g3 + 0);
    float py = __builtin_nontemporal_load(xyz + g3 + 1);
    float pz = __builtin_nontemporal_load(xyz + g3 + 2);

    v4f rq = __builtin_nontemporal_load((const v4f*)rotr + g);   // wxyz, 16B aligned

    float sx = __builtin_nontemporal_load(scal + g3 + 0);
    float sy = __builtin_nontemporal_load(scal + g3 + 1);
    float sz = __builtin_nontemporal_load(scal + g3 + 2);

    float f0 = __builtin_nontemporal_load(fdc + g3 + 0);
    float f1 = __builtin_nontemporal_load(fdc + g3 + 1);
    float f2 = __builtin_nontemporal_load(fdc + g3 + 2);

    float orw = __builtin_nontemporal_load(opac + g);

    // Face record (regular cached loads: table is hot in L2/L0).
    float r00 = fr[0], r01 = fr[1], r02 = fr[2];
    float r10 = fr[3], r11 = fr[4], r12 = fr[5];
    float r20 = fr[6], r21 = fr[7], r22 = fr[8];
    float ccx = fr[9], ccy = fr[10], ccz = fr[11];
    float fs  = fr[12];
    float fqw = fr[13], fqx = fr[14], fqy = fr[15], fqz = fr[16];

    // g_xyz = (R @ xyz) * fs + center
    float gx = (r00 * px + r01 * py + r02 * pz) * fs + ccx;
    float gy = (r10 * px + r11 * py + r12 * pz) * fs + ccy;
    float gz = (r20 * px + r21 * py + r22 * pz) * fs + ccz;

    // opacity / scales
    float op  = 1.0f / (1.0f + expf(-orw));
    float scx = expf(sx) * fs;
    float scy = expf(sy) * fs;
    float scz = expf(sz) * fs;

    // rot = rotation_raw / (||.|| + 1e-12), wxyz
    float rn  = sqrtf(rq.x * rq.x + rq.y * rq.y + rq.z * rq.z + rq.w * rq.w);
    float rin = 1.0f / (rn + 1e-12f);
    float qw = rq.x * rin, qx = rq.y * rin, qy = rq.z * rin, qz = rq.w * rin;

    // quaternion product (fq * rot) in wxyz
    float ow = fqw * qw - (fqx * qx + fqy * qy + fqz * qz);
    float ox = fqw * qx + qw * fqx + (fqy * qz - fqz * qy);
    float oy = fqw * qy + qw * fqy + (fqz * qx - fqx * qz);
    float oz = fqw * qz + qw * fqz + (fqx * qy - fqy * qx);
    float oin = 1.0f / sqrtf(ow * ow + ox * ox + oy * oy + oz * oz);
    ow *= oin; ox *= oin; oy *= oin; oz *= oin;

    // color = clip([0.5 + 0.282*shs0, opacity] * 255, 0, 255)
    float cr = fminf(fmaxf((0.5f + 0.282f * f0) * 255.0f, 0.0f), 255.0f);
    float cg = fminf(fmaxf((0.5f + 0.282f * f1) * 255.0f, 0.0f), 255.0f);
    float cb = fminf(fmaxf((0.5f + 0.282f * f2) * 255.0f, 0.0f), 255.0f);
    float ca = fminf(fmaxf(op * 255.0f, 0.0f), 255.0f);

    if (fullBlock) {
        // Stage the 14-float row in LDS (writes are bank-conflict-free:
        // 14*L mod 64 covers 32 distinct banks across a wave), then push the
        // wave's contiguous 1792B tile to HBM with async LDS->global stores.
        float* row = smem + tid * 14;
        row[0]  = gx;  row[1]  = gy;  row[2]  = gz;
        row[3]  = scx; row[4]  = scy; row[5]  = scz;
        row[6]  = cr;  row[7]  = cg;  row[8]  = cb;  row[9] = ca;
        row[10] = ow;  row[11] = ox;  row[12] = oy;  row[13] = oz;

        const unsigned wave = tid >> 5;
        const unsigned lane = tid & 31u;
        // Low 32 bits of a generic pointer to LDS are the wave-relative LDS
        // byte offset (aperture lives in the high 32 bits).
        const uint32_t lds0 = (uint32_t)(uintptr_t)(void*)smem + wave * 1792u;
        const uint64_t gl0  = (uint64_t)(uintptr_t)out
                            + (uint64_t)blockBase * 56u + (uint64_t)wave * 1792u;

        // Wave-local: our own DS writes must land before the async engine reads.
        asm volatile("s_wait_dscnt 0x0" ::: "memory");

        uint64_t ga = gl0 + (uint64_t)(lane * 16u);
        uint32_t la = lds0 + lane * 16u;
        asm volatile("global_store_async_from_lds_b128 %0, %1, off"
                     :: "v"(ga), "v"(la) : "memory");
        asm volatile("global_store_async_from_lds_b128 %0, %1, off"
                     :: "v"(ga + 512u), "v"(la + 512u) : "memory");
        asm volatile("global_store_async_from_lds_b128 %0, %1, off"
                     :: "v"(ga + 1024u), "v"(la + 1024u) : "memory");
        uint64_t gb = gl0 + 1536u + (uint64_t)(lane * 8u);
        uint32_t lb = lds0 + 1536u + lane * 8u;
        asm volatile("global_store_async_from_lds_b64 %0, %1, off"
                     :: "v"(gb), "v"(lb) : "memory");

        // Drain before the wave exits (S_ENDPGM also wait-idles; explicit for safety).
        asm volatile("s_wait_asynccnt 0x0" ::: "memory");
    } else {
        // Partial tail block: direct non-temporal 8B stores (row stride 56B).
        float* o = out + (size_t)g * 14;
        v2f t;
        t.x = gx;  t.y = gy;  __builtin_nontemporal_store(t, (v2f*)(o + 0));
        t.x = gz;  t.y = scx; __builtin_nontemporal_store(t, (v2f*)(o + 2));
        t.x = scy; t.y = scz; __builtin_nontemporal_store(t, (v2f*)(o + 4));
        t.x = cr;  t.y = cg;  __builtin_nontemporal_store(t, (v2f*)(o + 6));
        t.x = cb;  t.y = ca;  __builtin_nontemporal_store(t, (v2f*)(o + 8));
        t.x = ow;  t.y = ox;  __builtin_nontemporal_store(t, (v2f*)(o + 10));
        t.x = oy;  t.y = oz;  __builtin_nontemporal_store(t, (v2f*)(o + 12));
    }
}

// ---------------------------------------------------------------------------
extern "C" void kernel_launch(void* const* d_in, const int* in_sizes, int n_in,
                              void* d_out, int out_size, void* d_ws, size_t ws_size,
                              hipStream_t stream) {
    (void)n_in; (void)out_size; (void)ws_size;

    const float* verts   = (const float*)d_in[0];
    const float* xyz     = (const float*)d_in[1];
    const float* opac    = (const float*)d_in[2];
    const float* scal    = (const float*)d_in[3];
    const float* rotr    = (const float*)d_in[4];
    const float* fdc     = (const float*)d_in[5];
    // d_in[6] = features_rest: UNUSED by the reference output -- never touched.
    const int*   faces   = (const int*)d_in[7];
    const int*   binding = (const int*)d_in[8];

    const int n_faces = in_sizes[7] / 3;
    const int n_gauss = in_sizes[8];

    float* ftab = (float*)d_ws;   // n_faces * FACE_STRIDE floats (~780KB)

    dim3 blk(256);
    dim3 grid_f((n_faces + 255) / 256);
    dim3 grid_g((n_gauss + 255) / 256);

    face_table_kernel<<<grid_f, blk, 0, stream>>>(verts, faces, ftab, n_faces);
    gaussian_kernel<<<grid_g, blk, 0, stream>>>(xyz, opac, scal, rotr, fdc,
                                                binding, ftab, (float*)d_out, n_gauss);
}